// QG_88570815578617
// MI455X (gfx1250) — compile-verified
//
#include <hip/hip_runtime.h>

#define NQ      22
#define NSTATE  (1u << NQ)
#define NLAYERS 8
#define NGROUPS 6
#define TILES_PER_WAVE 8

typedef __attribute__((ext_vector_type(2))) float v2f;
typedef __attribute__((ext_vector_type(8))) float v8f;

// ---------------------------------------------------------------------------
// Build per-layer fused gate matrices.
// Per wire: U = RX(c) * RZ(b) * RY(a)  (2x2 complex, closed form).
// Per group g<5: G16 = U[4g] (x) U[4g+1] (x) U[4g+2] (x) U[4g+3]
// Group 5 (wires 20,21): G16 = I4 (x) (U20 (x) U21)   -> identity on field
// bits [3:2], gate on bits [1:0].
// Output layout per (layer,group): 512 floats = Gr[16][16] then Gi[16][16].
// ---------------------------------------------------------------------------
__global__ void build_gates(const float* __restrict__ params,
                            float* __restrict__ gates) {
    __shared__ float2 U[NQ][2][2];
    const int l = blockIdx.x / NGROUPS;
    const int g = blockIdx.x % NGROUPS;
    const int t = threadIdx.x;

    if (t < NQ) {
        const int w = t;
        const float a = params[l * 3 * NQ + w];            // RY angle
        const float b = params[l * 3 * NQ + NQ + w];       // RZ angle
        const float c = params[l * 3 * NQ + 2 * NQ + w];   // RX angle
        const float cy = cosf(0.5f * a), sy = sinf(0.5f * a);
        const float cb = cosf(0.5f * b), sb = sinf(0.5f * b);
        const float cx = cosf(0.5f * c), sx = sinf(0.5f * c);
        const float P1 = cx * cy, P2 = sx * sy, P3 = cx * sy, P4 = sx * cy;
        U[w][0][0] = make_float2( P1 * cb + P2 * sb, -(P1 * sb + P2 * cb));
        U[w][0][1] = make_float2(-P3 * cb + P4 * sb,   P3 * sb - P4 * cb);
        U[w][1][0] = make_float2( P3 * cb - P4 * sb,   P3 * sb - P4 * cb);
        U[w][1][1] = make_float2( P2 * sb + P1 * cb,   P2 * cb + P1 * sb);
    }
    __syncthreads();

    const int r = t >> 4, k = t & 15;
    float2 acc = make_float2(1.f, 0.f);
    if (g < 5) {
        #pragma unroll
        for (int j = 0; j < 4; ++j) {
            const int w = 4 * g + j;
            const float2 u = U[w][(r >> (3 - j)) & 1][(k >> (3 - j)) & 1];
            acc = make_float2(acc.x * u.x - acc.y * u.y,
                              acc.x * u.y + acc.y * u.x);
        }
    } else {
        if ((r >> 2) != (k >> 2)) {
            acc = make_float2(0.f, 0.f);
        } else {
            const float2 u0 = U[20][(r >> 1) & 1][(k >> 1) & 1];
            const float2 u1 = U[21][r & 1][k & 1];
            acc = make_float2(u0.x * u1.x - u0.y * u1.y,
                              u0.x * u1.y + u0.y * u1.x);
        }
    }
    float* base = gates + (size_t)blockIdx.x * 512;
    base[r * 16 + k]       = acc.x;   // Gr
    base[256 + r * 16 + k] = acc.y;   // Gi
}

// ---------------------------------------------------------------------------
// |0...0> initial state.
// ---------------------------------------------------------------------------
__global__ void init_state(float2* __restrict__ s) {
    const unsigned i = blockIdx.x * blockDim.x + threadIdx.x;
    s[i] = (i == 0) ? make_float2(1.f, 0.f) : make_float2(0.f, 0.f);
}

// ---------------------------------------------------------------------------
// Apply one 16x16 complex unitary on the 4-bit index field [B+3:B] via
// V_WMMA_F32_16X16X4_F32.  State viewed as S[16][2^18]; each wave computes
// 16x16 complex D-tiles:  Dr = Gr*Sr - Gi*Si,  Di = Gr*Si + Gi*Sr.
// fuse_perm: read through the CNOT-chain permutation  addr -> addr^(addr>>1)
// (Gray map = inverse of the prefix-XOR the CNOT ladder performs).
// ---------------------------------------------------------------------------
__global__ void __launch_bounds__(256)
apply_group(const float2* __restrict__ in, float2* __restrict__ out,
            const float* __restrict__ gmat, int B, int fuse_perm) {
    const unsigned tid  = blockIdx.x * blockDim.x + threadIdx.x;
    const unsigned wave = tid >> 5;
    const unsigned lane = threadIdx.x & 31u;
    const unsigned m    = lane & 15u;   // A row / B,D column
    const unsigned half = lane >> 4;    // selects K (A,B) / M+8 (D)

    const float* Gr = gmat;
    const float* Gi = gmat + 256;

    // Gate (A-matrix) fragments, loaded once per wave. ISA layout:
    // VGPR0: K = 2*half, VGPR1: K = 2*half+1 within each K=4 chunk.
    v2f Ar[4], Ai[4], AiN[4];
    #pragma unroll
    for (int kb = 0; kb < 4; ++kb) {
        const int k0 = 4 * kb + 2 * (int)half;
        Ar[kb]  = (v2f){ Gr[m * 16 + k0],  Gr[m * 16 + k0 + 1] };
        Ai[kb]  = (v2f){ Gi[m * 16 + k0],  Gi[m * 16 + k0 + 1] };
        AiN[kb] = (v2f){ -Ai[kb].x, -Ai[kb].y };
    }

    const unsigned loMask = (1u << B) - 1u;

    for (int tt = 0; tt < TILES_PER_WAVE; ++tt) {
        const unsigned tile = wave * TILES_PER_WAVE + (unsigned)tt;
        const unsigned col  = tile * 16u + m;          // this lane's column
        const unsigned lo   = col & loMask;
        const unsigned hi   = col >> B;
        const unsigned base = (hi << (B + 4)) | lo;    // amp index at field=0

        // B-matrix fragments (state columns), complex split into Re/Im.
        v2f Br[4], Bi[4];
        #pragma unroll
        for (int kb = 0; kb < 4; ++kb) {
            unsigned ke = base + ((4u * (unsigned)kb + 2u * half) << B);
            unsigned ko = ke + (1u << B);
            if (fuse_perm) { ke ^= (ke >> 1); ko ^= (ko >> 1); }
            const float2 se = in[ke];
            const float2 so = in[ko];
            Br[kb] = (v2f){ se.x, so.x };
            Bi[kb] = (v2f){ se.y, so.y };
        }

        v8f dr = {};
        v8f di = {};
        #pragma unroll
        for (int kb = 0; kb < 4; ++kb) {
            dr = __builtin_amdgcn_wmma_f32_16x16x4_f32(
                     false, Ar[kb],  false, Br[kb], (short)0, dr, false, false);
            dr = __builtin_amdgcn_wmma_f32_16x16x4_f32(
                     false, AiN[kb], false, Bi[kb], (short)0, dr, false, false);
            di = __builtin_amdgcn_wmma_f32_16x16x4_f32(
                     false, Ar[kb],  false, Bi[kb], (short)0, di, false, false);
            di = __builtin_amdgcn_wmma_f32_16x16x4_f32(
                     false, Ai[kb],  false, Br[kb], (short)0, di, false, false);
        }

        // D layout: VGPR v holds M = v + 8*half, N = lane&15.
        #pragma unroll
        for (int v = 0; v < 8; ++v) {
            const unsigned mm = (unsigned)v + 8u * half;
            out[base + (mm << B)] = make_float2(dr[v], di[v]);
        }
    }
}

// ---------------------------------------------------------------------------
// probs: apply the final layer's CNOT permutation via addressing, |amp|^2.
// ---------------------------------------------------------------------------
__global__ void probs_kernel(const float2* __restrict__ s,
                             float* __restrict__ out) {
    const unsigned i = blockIdx.x * blockDim.x + threadIdx.x;
    const unsigned j = i ^ (i >> 1);
    const float2 a = s[j];
    out[i] = a.x * a.x + a.y * a.y;
}

extern "C" void kernel_launch(void* const* d_in, const int* in_sizes, int n_in,
                              void* d_out, int out_size, void* d_ws, size_t ws_size,
                              hipStream_t stream) {
    const float* params = (const float*)d_in[0];

    char* wsb = (char*)d_ws;
    float* gates = (float*)wsb;                               // 48*512 floats
    float2* S0 = (float2*)(wsb + (4ull << 20));               // 32 MB
    float2* S1 = (float2*)(wsb + (4ull << 20) + (size_t)NSTATE * sizeof(float2));

    build_gates<<<NLAYERS * NGROUPS, 256, 0, stream>>>(params, gates);
    init_state<<<NSTATE / 256, 256, 0, stream>>>(S0);

    static const int Boff[NGROUPS] = {18, 14, 10, 6, 2, 0};
    float2* cur = S0;
    float2* nxt = S1;
    for (int l = 0; l < NLAYERS; ++l) {
        for (int g = 0; g < NGROUPS; ++g) {
            const float* gm = gates + (size_t)(l * NGROUPS + g) * 512;
            const int fuse = (g == 0 && l > 0) ? 1 : 0;
            // 256 blocks * 8 waves * 8 tiles * 16 cols = 2^18 columns exactly.
            apply_group<<<256, 256, 0, stream>>>(cur, nxt, gm, Boff[g], fuse);
            float2* tmp = cur; cur = nxt; nxt = tmp;
        }
    }
    // 48 passes (even) -> result back in S0 == cur.
    probs_kernel<<<NSTATE / 256, 256, 0, stream>>>(cur, (float*)d_out);
}